// GraphSAGE_69733089018244
// MI455X (gfx1250) — compile-verified
//
#include <hip/hip_runtime.h>

// ---------------------------------------------------------------------------
// GraphSAGE forward, MI455X (gfx1250, wave32).
//   Layers: 128->256 relu, 256->128 relu, 128->256 relu, 256->256.
//   neigh[d] += h[src]*inv_deg[d]  (edge aggregation, L2-resident atomics)
//   out = h @ Ws + neigh @ Wn + b  (fused dual-GEMM, V_WMMA_F32_16X16X4_F32,
//                                   weight tiles staged via async-to-LDS DMA)
// ---------------------------------------------------------------------------

typedef __attribute__((ext_vector_type(2))) float v2f;
typedef __attribute__((ext_vector_type(8))) float v8f;

#define N_NODES 50000
#define N_EDGES 800000

#if defined(__has_builtin)
#  if __has_builtin(__builtin_amdgcn_global_load_async_to_lds_b128)
#    define HAVE_ASYNC_LDS 1
#  endif
#endif
#ifndef HAVE_ASYNC_LDS
#  define HAVE_ASYNC_LDS 0
#endif

#if HAVE_ASYNC_LDS
// builtin expects: (v4i addrspace(1)*, v4i addrspace(3)*, imm offset, imm cpol)
typedef int v4i_g __attribute__((vector_size(16)));
typedef __attribute__((address_space(1))) v4i_g* as1_v4i_p;
typedef __attribute__((address_space(3))) v4i_g* as3_v4i_p;
#endif

__device__ __forceinline__ void wait_async0() {
#if HAVE_ASYNC_LDS
#  if __has_builtin(__builtin_amdgcn_s_wait_asynccnt)
    __builtin_amdgcn_s_wait_asynccnt(0);
#  else
    asm volatile("s_wait_asynccnt 0x0" ::: "memory");
#  endif
#endif
}

// ------------------------------- utility -----------------------------------

__global__ void zero_f32_vec4(float* __restrict__ p, long long n4) {
    long long i = (long long)blockIdx.x * blockDim.x + threadIdx.x;
    if (i < n4) ((float4*)p)[i] = make_float4(0.f, 0.f, 0.f, 0.f);
}

__global__ void degree_kernel(const int* __restrict__ dst, float* __restrict__ deg, int e) {
    int i = blockIdx.x * blockDim.x + threadIdx.x;
    if (i < e) {
        __hip_atomic_fetch_add(&deg[dst[i]], 1.0f,
                               __ATOMIC_RELAXED, __HIP_MEMORY_SCOPE_AGENT);
    }
}

__global__ void invdeg_kernel(float* __restrict__ deg, int n) {
    int i = blockIdx.x * blockDim.x + threadIdx.x;
    if (i < n) deg[i] = 1.0f / fmaxf(deg[i], 1.0f);
}

// --------------------------- edge aggregation ------------------------------
// One thread per (edge, float4-chunk). inv_deg[dst] is folded into the edge
// contribution (scaling commutes with the segment sum), so downstream GEMM
// tiles are pure memory copies. h, neigh, inv_deg are all L2-resident on
// MI455X (192 MB L2), so gathers and fp32 atomics resolve at L2 bandwidth.

__global__ void aggregate_kernel(const float* __restrict__ h,
                                 const int* __restrict__ src,
                                 const int* __restrict__ dst,
                                 const float* __restrict__ inv_deg,
                                 float* __restrict__ neigh,
                                 int shift /* log2(dim/4) */, int dim) {
    long long total = (long long)N_EDGES << shift;
    long long i = (long long)blockIdx.x * blockDim.x + threadIdx.x;
    if (i >= total) return;
    int e = (int)(i >> shift);
    int f = (int)(i & ((1 << shift) - 1));
    int s = src[e];
    int d = dst[e];
    float w = inv_deg[d];
    float4 v = ((const float4*)(h + (size_t)s * dim))[f];
    float* p = neigh + (size_t)d * dim + 4 * (size_t)f;
    __hip_atomic_fetch_add(p + 0, v.x * w, __ATOMIC_RELAXED, __HIP_MEMORY_SCOPE_AGENT);
    __hip_atomic_fetch_add(p + 1, v.y * w, __ATOMIC_RELAXED, __HIP_MEMORY_SCOPE_AGENT);
    __hip_atomic_fetch_add(p + 2, v.z * w, __ATOMIC_RELAXED, __HIP_MEMORY_SCOPE_AGENT);
    __hip_atomic_fetch_add(p + 3, v.w * w, __ATOMIC_RELAXED, __HIP_MEMORY_SCOPE_AGENT);
}

// ------------------------- fused dual-GEMM (WMMA) --------------------------
// out[M x dout] = hself @ Wself + hneigh @ Wneigh + bias, optional relu.
// Block: 256 threads = 8 wave32, tile 64x64; wave owns 16x32 (two 16x16 C
// fragments). Weight tiles are double-buffered in LDS and filled with
// GLOBAL_LOAD_ASYNC_TO_LDS_B128 (next chunk's DMA issued before the WMMA
// burst, completion enforced by s_wait_asynccnt + barrier).

#define BM 64
#define BN 64
#define BK 16

__launch_bounds__(256)
__global__ void sage_gemm_kernel(const float* __restrict__ hself,
                                 const float* __restrict__ hneigh,
                                 const float* __restrict__ Wself,
                                 const float* __restrict__ Wneigh,
                                 const float* __restrict__ bias,
                                 float* __restrict__ out,
                                 int nrows, int din, int dout, int do_relu) {
    __shared__ float As[BM][BK + 1];      // +1 pad: 17 coprime 64 banks
    __shared__ float An[BM][BK + 1];
    __shared__ float Bs[2][BK][BN];       // double-buffered async-DMA targets
    __shared__ float Bn[2][BK][BN];

    const int t    = threadIdx.x;
    const int lane = t & 31;
    const int wave = t >> 5;
    const int wm   = wave & 3;   // M subtile 0..3 (16 rows each)
    const int wn2  = wave >> 2;  // N half 0..1 (32 cols each)
    const int lrow = lane & 15;
    const int lhi  = lane >> 4;  // K/row-half selector per ISA layout

    const int bm  = blockIdx.y * BM;
    const int bn0 = blockIdx.x * BN;

    // cooperative-load coordinates
    const int ar = t >> 2;            // 0..63 (A row within tile)
    const int ac = (t & 3) << 2;      // 0,4,8,12 (A col, float4)
    const int br = t >> 4;            // 0..15 (B row within tile)
    const int bc = (t & 15) << 2;     // 0..60 (B col, float4)

    const int  arow = bm + ar;
    const bool aval = arow < nrows;

    v8f acc0 = {};
    v8f acc1 = {};

    auto copy_w_tile = [&](const float* __restrict__ W, float (&dst)[BK][BN], int k0) {
        const size_t wo = (size_t)(k0 + br) * dout + bn0 + bc;
#if HAVE_ASYNC_LDS
        __builtin_amdgcn_global_load_async_to_lds_b128(
            (as1_v4i_p)(W + wo), (as3_v4i_p)(&dst[br][bc]), 0, 0);
#else
        *(float4*)&dst[br][bc] = *(const float4*)(W + wo);
#endif
    };

    const int nch = din / BK;

    // prologue: DMA chunk 0 weight tiles into buffer 0
    copy_w_tile(Wself,  Bs[0], 0);
    copy_w_tile(Wneigh, Bn[0], 0);

    for (int c = 0; c < nch; ++c) {
        const int k0  = c * BK;
        const int cur = c & 1;
        const int nxt = cur ^ 1;

        // A tiles via VGPRs (row-guarded), padded LDS
        float4 a4 = make_float4(0.f, 0.f, 0.f, 0.f);
        float4 n4 = a4;
        if (aval) {
            const size_t ro = (size_t)arow * din + k0 + ac;
            a4 = *(const float4*)(hself + ro);
            n4 = *(const float4*)(hneigh + ro);
            if (k0 + BK < din) {  // warm L2/L0 for next chunk
                __builtin_prefetch(hself + ro + BK, 0, 3);
                __builtin_prefetch(hneigh + ro + BK, 0, 3);
            }
        }
        As[ar][ac + 0] = a4.x; As[ar][ac + 1] = a4.y;
        As[ar][ac + 2] = a4.z; As[ar][ac + 3] = a4.w;
        An[ar][ac + 0] = n4.x; An[ar][ac + 1] = n4.y;
        An[ar][ac + 2] = n4.z; An[ar][ac + 3] = n4.w;

        wait_async0();     // this chunk's weight DMA has landed in LDS
        __syncthreads();   // all waves: A stores + B DMA visible

        if (c + 1 < nch) { // overlap next chunk's DMA with the WMMA burst
            copy_w_tile(Wself,  Bs[nxt], k0 + BK);
            copy_w_tile(Wneigh, Bn[nxt], k0 + BK);
        }

        const int am = wm * 16 + lrow;
        const int nb = wn2 * 32 + lrow;
#pragma unroll
        for (int kk = 0; kk < BK; kk += 4) {
            const int ka = kk + 2 * lhi;  // lanes 0-15: K=kk,kk+1; 16-31: kk+2,kk+3
            v2f aS, aN, b0s, b1s, b0n, b1n;
            aS[0]  = As[am][ka];            aS[1]  = As[am][ka + 1];
            aN[0]  = An[am][ka];            aN[1]  = An[am][ka + 1];
            b0s[0] = Bs[cur][ka][nb];       b0s[1] = Bs[cur][ka + 1][nb];
            b1s[0] = Bs[cur][ka][nb + 16];  b1s[1] = Bs[cur][ka + 1][nb + 16];
            b0n[0] = Bn[cur][ka][nb];       b0n[1] = Bn[cur][ka + 1][nb];
            b1n[0] = Bn[cur][ka][nb + 16];  b1n[1] = Bn[cur][ka + 1][nb + 16];

            acc0 = __builtin_amdgcn_wmma_f32_16x16x4_f32(false, aS, false, b0s,
                                                         (short)0, acc0, false, false);
            acc0 = __builtin_amdgcn_wmma_f32_16x16x4_f32(false, aN, false, b0n,
                                                         (short)0, acc0, false, false);
            acc1 = __builtin_amdgcn_wmma_f32_16x16x4_f32(false, aS, false, b1s,
                                                         (short)0, acc1, false, false);
            acc1 = __builtin_amdgcn_wmma_f32_16x16x4_f32(false, aN, false, b1n,
                                                         (short)0, acc1, false, false);
        }
        __syncthreads();   // compute done before A tiles / next buffer reuse
    }

    // Epilogue: C/D layout — VGPR i holds row (i + 8*lhi), col = lrow.
    const int col0 = bn0 + wn2 * 32 + lrow;
    const float bia0 = bias[col0];
    const float bia1 = bias[col0 + 16];
#pragma unroll
    for (int i = 0; i < 8; ++i) {
        const int row = bm + wm * 16 + i + 8 * lhi;
        if (row < nrows) {
            float v0 = acc0[i] + bia0;
            float v1 = acc1[i] + bia1;
            if (do_relu) { v0 = fmaxf(v0, 0.f); v1 = fmaxf(v1, 0.f); }
            out[(size_t)row * dout + col0]      = v0;
            out[(size_t)row * dout + col0 + 16] = v1;
        }
    }
}

// ------------------------------- launch ------------------------------------

extern "C" void kernel_launch(void* const* d_in, const int* in_sizes, int n_in,
                              void* d_out, int out_size, void* d_ws, size_t ws_size,
                              hipStream_t stream) {
    const float* feat = (const float*)d_in[0];
    const int*   src  = (const int*)d_in[1];
    const int*   dst  = (const int*)d_in[2];
    const float* w1s = (const float*)d_in[3];
    const float* w1n = (const float*)d_in[4];
    const float* b1  = (const float*)d_in[5];
    const float* w2s = (const float*)d_in[6];
    const float* w2n = (const float*)d_in[7];
    const float* b2  = (const float*)d_in[8];
    const float* w3s = (const float*)d_in[9];
    const float* w3n = (const float*)d_in[10];
    const float* b3  = (const float*)d_in[11];
    const float* w4s = (const float*)d_in[12];
    const float* w4n = (const float*)d_in[13];
    const float* b4  = (const float*)d_in[14];

    // workspace layout (floats): [inv_deg | bufA | bufB | neigh]
    float* wsf  = (float*)d_ws;
    float* deg  = wsf;                                   // N_NODES
    float* bufA = deg  + (size_t)N_NODES;                // N_NODES*256
    float* bufB = bufA + (size_t)N_NODES * 256;          // N_NODES*256
    float* nbuf = bufB + (size_t)N_NODES * 256;          // N_NODES*256

    // ---- degrees -> inv_deg (in place) ----
    {
        long long n4 = N_NODES / 4;
        zero_f32_vec4<<<(unsigned)((n4 + 255) / 256), 256, 0, stream>>>(deg, n4);
        degree_kernel<<<(N_EDGES + 255) / 256, 256, 0, stream>>>(dst, deg, N_EDGES);
        invdeg_kernel<<<(N_NODES + 255) / 256, 256, 0, stream>>>(deg, N_NODES);
    }

    auto run_layer = [&](const float* hin, int din, int dout,
                         const float* Ws, const float* Wn, const float* B,
                         float* hout, int relu) {
        long long n4 = (long long)N_NODES * din / 4;
        zero_f32_vec4<<<(unsigned)((n4 + 255) / 256), 256, 0, stream>>>(nbuf, n4);
        int shift = (din == 256) ? 6 : 5;  // log2(din/4)
        long long tot = (long long)N_EDGES << shift;
        aggregate_kernel<<<(unsigned)((tot + 255) / 256), 256, 0, stream>>>(
            hin, src, dst, deg, nbuf, shift, din);
        dim3 g(dout / BN, (N_NODES + BM - 1) / BM);
        sage_gemm_kernel<<<g, 256, 0, stream>>>(hin, nbuf, Ws, Wn, B,
                                                hout, N_NODES, din, dout, relu);
    };

    run_layer(feat, 128, 256, w1s, w1n, b1, bufA, 1);
    run_layer(bufA, 256, 128, w2s, w2n, b2, bufB, 1);
    run_layer(bufB, 128, 256, w3s, w3n, b3, bufA, 1);
    run_layer(bufA, 256, 256, w4s, w4n, b4, (float*)d_out, 0);
}